// GCN_8108898255052
// MI455X (gfx1250) — compile-verified
//
#include <hip/hip_runtime.h>
#include <hip/hip_bf16.h>

typedef __attribute__((ext_vector_type(2))) float v2f;
typedef __attribute__((ext_vector_type(8))) float v8f;

// ---------------------------------------------------------------------------
// Zero norm accumulators and the output/aggregation buffer (float4 stores).
// ---------------------------------------------------------------------------
__global__ void gcn_zero(float* __restrict__ norms, long n_norm4,
                         float* __restrict__ out, long n_out4) {
    long i      = (long)blockIdx.x * blockDim.x + threadIdx.x;
    long stride = (long)gridDim.x * blockDim.x;
    float4 z = make_float4(0.f, 0.f, 0.f, 0.f);
    for (long j = i; j < n_norm4; j += stride) ((float4*)norms)[j] = z;
    for (long j = i; j < n_out4;  j += stride) ((float4*)out)[j]   = z;
}

// ---------------------------------------------------------------------------
// Degree counting: atomic float adds into deg_out[src], deg_in[dst].
// ---------------------------------------------------------------------------
__global__ void gcn_degrees(const int* __restrict__ src, const int* __restrict__ dst,
                            float* __restrict__ deg_out, float* __restrict__ deg_in,
                            int n_edges) {
    long i      = (long)blockIdx.x * blockDim.x + threadIdx.x;
    long stride = (long)gridDim.x * blockDim.x;
    for (long e = i; e < n_edges; e += stride) {
        atomicAdd(&deg_out[src[e]], 1.0f);
        atomicAdd(&deg_in[dst[e]],  1.0f);
    }
}

// deg -> deg^{-1/2} (0 where deg == 0), applied to both arrays back to back.
__global__ void gcn_norms(float* __restrict__ deg, int n2) {
    int i = blockIdx.x * blockDim.x + threadIdx.x;
    if (i < n2) {
        float d = deg[i];
        deg[i] = (d > 0.f) ? rsqrtf(d) : 0.f;
    }
}

// ---------------------------------------------------------------------------
// out[i, 0:64] = norm[i] * (A[i, 0:K] @ W[0:K, 0:64])   via V_WMMA_F32_16X16X4_F32
//
// Block = 256 threads = 8 waves; wave w computes rows [row0, row0+16) x 64 cols
// with four 16x16 f32 accumulators.
//
// W is staged in LDS PAIR-INTERLEAVED:  ldsB[(k/2)*128 + 2n + (k&1)] = W[k][n]
// so a B fragment {W[ka][n], W[ka+1][n]} is a single aligned ds_load_b64, and
// the four N-tiles are the same base with immediate offsets 0/128/256/384 B.
//
// Fragment layouts (ISA 7.12.2, 32-bit, wave32):
//   A 16x4 : lanes 0-15 -> M=lane, VGPR0=K0,VGPR1=K1 ; lanes 16-31 -> K2,K3
//   B 4x16 : lanes 0-15 -> K0/K1 in v0/v1, N=lane    ; lanes 16-31 -> K2/K3
//   D 16x16: VGPR v -> M = v + 8*(lane>=16), N = lane&15
// Requires n_nodes % 16 == 0 (true for the reference: 100000).
// ---------------------------------------------------------------------------
__global__ void gcn_gemm_wmma(const float* __restrict__ A, const float* __restrict__ W,
                              const float* __restrict__ norm, float* __restrict__ out,
                              int n_nodes, int K) {
    extern __shared__ float ldsB[];   // (K/2) pair-rows x 128 floats

    // Cooperative pair-interleaved staging of W (K x 64, row-major in global).
    const int nW = K << 6;
    for (int i = threadIdx.x; i < nW; i += blockDim.x) {
        const int k = i >> 6;
        const int n = i & 63;
        ldsB[((k >> 1) << 7) + (n << 1) + (k & 1)] = W[i];
    }
    __syncthreads();

    const int wave = threadIdx.x >> 5;
    const int lane = threadIdx.x & 31;
    const int row0 = ((blockIdx.x << 3) + wave) << 4;   // 8 waves * 16 rows
    if (row0 >= n_nodes) return;

    const int half = lane >> 4;    // 0: K pair {0,1}, 1: K pair {2,3}
    const int l16  = lane & 15;
    const int rowA = row0 + l16;
    const float s  = norm[rowA];

    const float* __restrict__ arow = A + (long)rowA * K;

    v8f acc0 = {}, acc1 = {}, acc2 = {}, acc3 = {};

    for (int k = 0; k < K; k += 4) {
        const int ka = k + (half << 1);

        v2f a = *(const v2f*)(arow + ka);      // global_load_b64
        a.x *= s;
        a.y *= s;                              // v_pk_mul_f32

        const float* __restrict__ bp = ldsB + ((ka >> 1) << 7) + (l16 << 1);
        v2f b0 = *(const v2f*)(bp);            // ds_load_b64, offset 0
        v2f b1 = *(const v2f*)(bp + 32);       // ds_load_b64, offset 128
        v2f b2 = *(const v2f*)(bp + 64);       // ds_load_b64, offset 256
        v2f b3 = *(const v2f*)(bp + 96);       // ds_load_b64, offset 384

        acc0 = __builtin_amdgcn_wmma_f32_16x16x4_f32(false, a, false, b0,
                                                     (short)0, acc0, false, false);
        acc1 = __builtin_amdgcn_wmma_f32_16x16x4_f32(false, a, false, b1,
                                                     (short)0, acc1, false, false);
        acc2 = __builtin_amdgcn_wmma_f32_16x16x4_f32(false, a, false, b2,
                                                     (short)0, acc2, false, false);
        acc3 = __builtin_amdgcn_wmma_f32_16x16x4_f32(false, a, false, b3,
                                                     (short)0, acc3, false, false);
    }

    float* __restrict__ o = out + (long)row0 * 64;
#pragma unroll
    for (int v = 0; v < 8; ++v) {
        const long m = (long)(v + (half << 3)) * 64;
        o[m + 0  + l16] = acc0[v];
        o[m + 16 + l16] = acc1[v];
        o[m + 32 + l16] = acc2[v];
        o[m + 48 + l16] = acc3[v];
    }
}

// ---------------------------------------------------------------------------
// Edge scatter: agg[dst[e], :] += h[src[e], :]. 16 threads per edge, each
// moving a float4 (coalesced 256B gather per edge group, fp32 atomics to L2).
// ---------------------------------------------------------------------------
__global__ void gcn_scatter(const float* __restrict__ h, const int* __restrict__ src,
                            const int* __restrict__ dst, float* __restrict__ agg,
                            int n_edges) {
    long total  = (long)n_edges << 4;
    long i      = (long)blockIdx.x * blockDim.x + threadIdx.x;
    long stride = (long)gridDim.x * blockDim.x;
    for (long t = i; t < total; t += stride) {
        const int  e = (int)(t >> 4);
        const int  f = (int)(t & 15) << 2;
        const int  se = src[e];
        const int  de = dst[e];
        float4 v = *(const float4*)(h + (long)se * 64 + f);
        float* o = agg + (long)de * 64 + f;
        atomicAdd(o + 0, v.x);
        atomicAdd(o + 1, v.y);
        atomicAdd(o + 2, v.z);
        atomicAdd(o + 3, v.w);
    }
}

// h1 = relu(agg * norm_in + b1); re-zero agg (d_out) for reuse by layer 2.
__global__ void gcn_post1(float* __restrict__ agg, const float* __restrict__ norm_in,
                          const float* __restrict__ bias, float* __restrict__ h1,
                          long n64) {
    long i      = (long)blockIdx.x * blockDim.x + threadIdx.x;
    long stride = (long)gridDim.x * blockDim.x;
    for (long t = i; t < n64; t += stride) {
        float v = agg[t];
        agg[t] = 0.f;
        float y = fmaf(v, norm_in[t >> 6], bias[t & 63]);
        h1[t] = (y > 0.f) ? y : 0.f;
    }
}

// out = agg * norm_in + b2 (in place on d_out).
__global__ void gcn_post2(float* __restrict__ agg, const float* __restrict__ norm_in,
                          const float* __restrict__ bias, long n64) {
    long i      = (long)blockIdx.x * blockDim.x + threadIdx.x;
    long stride = (long)gridDim.x * blockDim.x;
    for (long t = i; t < n64; t += stride)
        agg[t] = fmaf(agg[t], norm_in[t >> 6], bias[t & 63]);
}

// ---------------------------------------------------------------------------
extern "C" void kernel_launch(void* const* d_in, const int* in_sizes, int n_in,
                              void* d_out, int out_size, void* d_ws, size_t ws_size,
                              hipStream_t stream) {
    const float* x  = (const float*)d_in[0];
    const int*   src = (const int*)d_in[1];
    const int*   dst = (const int*)d_in[2];
    const float* W1 = (const float*)d_in[3];
    const float* b1 = (const float*)d_in[4];
    const float* W2 = (const float*)d_in[5];
    const float* b2 = (const float*)d_in[6];
    float* out = (float*)d_out;

    const int hidden   = in_sizes[4];            // 64
    const int in_feats = in_sizes[3] / hidden;   // 128
    const int n_nodes  = in_sizes[0] / in_feats; // 100000
    const int n_edges  = in_sizes[1];            // 1600000

    // Workspace layout: [norm_out N][norm_in N][bufA N*64]  (~26.4 MB)
    float* norm_out = (float*)d_ws;
    float* norm_in  = norm_out + n_nodes;
    float* bufA     = norm_in  + n_nodes;

    const long n64 = (long)n_nodes * hidden;

    gcn_zero<<<2048, 256, 0, stream>>>(norm_out, (2L * n_nodes) >> 2, out, n64 >> 2);
    gcn_degrees<<<4096, 256, 0, stream>>>(src, dst, norm_out, norm_in, n_edges);
    gcn_norms<<<(2 * n_nodes + 255) / 256, 256, 0, stream>>>(norm_out, 2 * n_nodes);

    // Layer 1: bufA = (x * norm_out) @ W1
    gcn_gemm_wmma<<<(n_nodes + 127) / 128, 256,
                    (size_t)in_feats * hidden * sizeof(float), stream>>>(
        x, W1, norm_out, bufA, n_nodes, in_feats);
    gcn_scatter<<<8192, 256, 0, stream>>>(bufA, src, dst, out, n_edges);
    gcn_post1<<<8192, 256, 0, stream>>>(out, norm_in, b1, bufA, n64);

    // Layer 2: bufA = (relu(h1) * norm_out) @ W2  (in place; per-row tiles)
    gcn_gemm_wmma<<<(n_nodes + 127) / 128, 256,
                    (size_t)hidden * hidden * sizeof(float), stream>>>(
        bufA, W2, norm_out, bufA, n_nodes, hidden);
    gcn_scatter<<<8192, 256, 0, stream>>>(bufA, src, dst, out, n_edges);
    gcn_post2<<<8192, 256, 0, stream>>>(out, norm_in, b2, n64);
}